// Salsa20Cipher_30580167148127
// MI455X (gfx1250) — compile-verified
//
#include <hip/hip_runtime.h>
#include <stdint.h>

// ---------------------------------------------------------------------------
// Soft-Salsa20 for MI455X (gfx1250).
// Transcendental-bound: ~1008 sigmoids/element -> one v_tanh_f32 each
// (sigmoid(z) = 0.5 + 0.5*tanh(z/2), half the TRANS ops of exp+rcp).
// rotate_left collapses to mul/fract/fma/fract because every 2^n scale is an
// exact power-of-two multiply; v_fract_f32 fuses floor+sub and gives the
// Python-mod behavior for slightly-negative inputs.
// Plaintext is staged via CDNA5 async global->LDS DMA, overlapped with the
// ~4000-cycle round computation (frees 16 VGPRs for occupancy).
// ---------------------------------------------------------------------------

#define F_TWO32 4294967296.0f
#define F_INV32 (1.0f / 4294967296.0f)

__device__ __forceinline__ float hw_tanh(float x) {
#if __has_builtin(__builtin_amdgcn_tanhf)
    return __builtin_amdgcn_tanhf(x);
#elif __has_builtin(__builtin_amdgcn_tanh_f32)
    return __builtin_amdgcn_tanh_f32(x);
#else
    float r;
    asm("v_tanh_f32 %0, %1" : "=v"(r) : "v"(x));
    return r;
#endif
}

__device__ __forceinline__ float hw_fract(float x) {
#if __has_builtin(__builtin_amdgcn_fractf)
    return __builtin_amdgcn_fractf(x);   // v_fract_f32: x - floor(x)
#else
    return x - floorf(x);
#endif
}

// sigmoid(10*(v-0.5)) = 0.5 + 0.5*tanh(5v - 2.5)
__device__ __forceinline__ float sig_half(float v) {
    return __builtin_fmaf(0.5f, hw_tanh(__builtin_fmaf(5.0f, v, -2.5f)), 0.5f);
}

// soft_add: s - sigmoid(10*(s-1)) = (s-0.5) - 0.5*tanh(5s-5)     4 VALU + 1 tanh
__device__ __forceinline__ float soft_add(float x, float y) {
    float s = x + y;
    float t = hw_tanh(__builtin_fmaf(5.0f, s, -5.0f));
    return __builtin_fmaf(-0.5f, t, s - 0.5f);
}

// soft_xor: r = 1-(1-t1)(1-t2), 1-t1 = fma(xs,ys,1-xs), 1-t2 = fma(xs,ys,1-ys)
// 10 VALU + 2 tanh
__device__ __forceinline__ float soft_xor(float x, float y) {
    float xs = sig_half(x);
    float ys = sig_half(y);
    float m1 = __builtin_fmaf(xs, ys, 1.0f - xs);   // 1 - xs(1-ys)
    float m2 = __builtin_fmaf(xs, ys, 1.0f - ys);   // 1 - (1-xs)ys
    float r  = __builtin_fmaf(-m1, m2, 1.0f);
    return __builtin_amdgcn_fmed3f(r, 0.0f, 1.0f);  // clip [0,1]
}

// rotate_left(x, n):
//   t = x*2^n (exact);  result = frac( frac(t) + t*2^-32 )
// Matches the reference's mod-arithmetic: all 2^k scalings are exact, the
// single rounding point is the one addition, fused here into the fma.
// 4 VALU.
__device__ __forceinline__ float rotl(float x, float p2n) {
    float t  = x * p2n;
    float fr = hw_fract(t);
    float z  = __builtin_fmaf(t, F_INV32, fr);
    return hw_fract(z);
}

__device__ __forceinline__ void qround(float& a, float& b, float& c, float& d) {
    b = soft_xor(b, rotl(soft_add(a, d), 128.0f));     // n=7
    c = soft_xor(c, rotl(soft_add(b, a), 512.0f));     // n=9
    d = soft_xor(d, rotl(soft_add(c, b), 8192.0f));    // n=13
    a = soft_xor(a, rotl(soft_add(d, c), 262144.0f));  // n=18
}

__global__ __launch_bounds__(256) void salsa20_soft_kernel(
        const float* __restrict__ pt, const float* __restrict__ key,
        const float* __restrict__ nonce, float* __restrict__ out, int B) {
    __shared__ float pt_s[256 * 16];   // 16 KB: per-thread 64B plaintext slab

    const int tid = threadIdx.x;
    const int idx = blockIdx.x * 256 + tid;
    if (idx >= B) return;

    // ---- CDNA5 async DMA: plaintext -> LDS, overlapped with the rounds ----
    // Per-lane 64 bytes via 4x async b128; INST_OFFSET applies to both the
    // LDS and global addresses, so one (lds,voff) pair serves all four.
    uint32_t ldsa = (uint32_t)(uintptr_t)(void*)&pt_s[tid * 16];
    uint32_t goff = (uint32_t)idx * 64u;
    asm volatile("global_load_async_to_lds_b128 %0, %1, %2 offset:0"
                 :: "v"(ldsa), "v"(goff), "s"(pt) : "memory");
    asm volatile("global_load_async_to_lds_b128 %0, %1, %2 offset:16"
                 :: "v"(ldsa), "v"(goff), "s"(pt) : "memory");
    asm volatile("global_load_async_to_lds_b128 %0, %1, %2 offset:32"
                 :: "v"(ldsa), "v"(goff), "s"(pt) : "memory");
    asm volatile("global_load_async_to_lds_b128 %0, %1, %2 offset:48"
                 :: "v"(ldsa), "v"(goff), "s"(pt) : "memory");

    // ---- key / nonce (vectorized 128b / 64b loads) ----
    const float4 ka = *(const float4*)(key + (size_t)idx * 8);
    const float4 kb = *(const float4*)(key + (size_t)idx * 8 + 4);
    const float2 nn = *(const float2*)(nonce + (size_t)idx * 2);

    const float c0 = 1634760805.0f / F_TWO32;
    const float c1 =  857760878.0f / F_TWO32;
    const float c2 = 2036477234.0f / F_TWO32;
    const float c3 = 1797285236.0f / F_TWO32;

    float w[16];
    w[0]  = c0;    w[1]  = ka.x;  w[2]  = ka.y;  w[3]  = ka.z;
    w[4]  = ka.w;  w[5]  = c1;    w[6]  = nn.x;  w[7]  = nn.y;
    w[8]  = 0.0f;  w[9]  = 0.0f;  w[10] = c2;    w[11] = kb.x;
    w[12] = kb.y;  w[13] = kb.z;  w[14] = kb.w;  w[15] = c3;

    // 10 double-rounds, rolled (body ~600 instr keeps I-cache happy); the 4
    // column-QRs (and 4 row-QRs) are independent -> 4 chains of ILP to hide
    // v_tanh_f32 latency.
#pragma unroll 1
    for (int r = 0; r < 10; ++r) {
        qround(w[0],  w[4],  w[8],  w[12]);
        qround(w[5],  w[9],  w[13], w[1]);
        qround(w[10], w[14], w[2],  w[6]);
        qround(w[15], w[3],  w[7],  w[11]);
        qround(w[0],  w[1],  w[2],  w[3]);
        qround(w[5],  w[6],  w[7],  w[4]);
        qround(w[10], w[11], w[8],  w[9]);
        qround(w[15], w[12], w[13], w[14]);
    }

    // keystream = soft_add(w, initial_state)
    w[0]  = soft_add(w[0],  c0);    w[1]  = soft_add(w[1],  ka.x);
    w[2]  = soft_add(w[2],  ka.y);  w[3]  = soft_add(w[3],  ka.z);
    w[4]  = soft_add(w[4],  ka.w);  w[5]  = soft_add(w[5],  c1);
    w[6]  = soft_add(w[6],  nn.x);  w[7]  = soft_add(w[7],  nn.y);
    w[8]  = soft_add(w[8],  0.0f);  w[9]  = soft_add(w[9],  0.0f);
    w[10] = soft_add(w[10], c2);    w[11] = soft_add(w[11], kb.x);
    w[12] = soft_add(w[12], kb.y);  w[13] = soft_add(w[13], kb.z);
    w[14] = soft_add(w[14], kb.w);  w[15] = soft_add(w[15], c3);

    // DMA finished long ago (had ~4000 cycles of cover). Own-wave ASYNCcnt
    // ordering suffices: each thread reads only bytes its own lane's async
    // loads wrote, so no workgroup barrier is needed.
    asm volatile("s_wait_asynccnt 0" ::: "memory");

    const float* p = &pt_s[tid * 16];
    float4* o4 = (float4*)(out + (size_t)idx * 16);
#pragma unroll
    for (int g = 0; g < 4; ++g) {
        float4 o;
        o.x = soft_xor(p[g * 4 + 0], w[g * 4 + 0]);
        o.y = soft_xor(p[g * 4 + 1], w[g * 4 + 1]);
        o.z = soft_xor(p[g * 4 + 2], w[g * 4 + 2]);
        o.w = soft_xor(p[g * 4 + 3], w[g * 4 + 3]);
        o4[g] = o;
    }
}

extern "C" void kernel_launch(void* const* d_in, const int* in_sizes, int n_in,
                              void* d_out, int out_size, void* d_ws, size_t ws_size,
                              hipStream_t stream) {
    const float* pt    = (const float*)d_in[0];
    const float* key   = (const float*)d_in[1];
    const float* nonce = (const float*)d_in[2];
    float* out = (float*)d_out;

    const int B = in_sizes[0] / 16;            // 1048576
    const int blocks = (B + 255) / 256;
    salsa20_soft_kernel<<<blocks, 256, 0, stream>>>(pt, key, nonce, out, B);
}